// MultiHot_Embedding_74062416052471
// MI455X (gfx1250) — compile-verified
//
#include <hip/hip_runtime.h>
#include <hip/hip_bf16.h>

typedef __attribute__((ext_vector_type(2))) float v2f;
typedef __attribute__((ext_vector_type(4))) float v4f;
typedef __attribute__((ext_vector_type(8))) float v8f;

#define NUM_FEATURE 26
#define EMB_SIZE    20
#define BINS        50
#define CHANNELS    300   // 3*TOTAL

// ---------------------------------------------------------------------------
// Kernel 1: E[50][20] = multiHot_bin[50][300] @ W[300][20], exact f32 GEMM
// via V_WMMA_F32_16X16X4_F32. One block, 8 waves; wave w handles output tile
// (mt = w%4, nt = w/4). 75 K-steps of 4. Cost is negligible (~600 WMMA ops)
// but keeps the table build faithful to whatever multiHot_bin contains.
//
// VGPR layouts (ISA 7.12.2, wave32):
//   A (16x4 f32):  lane L -> M = L%16 ; VGPR j -> K = 2*(L/16) + j
//   B (4x16 f32):  lane L -> N = L%16 ; VGPR j -> K = 2*(L/16) + j
//   C (16x16 f32): VGPR r -> M = 8*(L/16) + r ; N = L%16
// ---------------------------------------------------------------------------
__global__ void mh_emb_build_table(const float* __restrict__ MH,  // [50][300]
                                   const float* __restrict__ W,   // [300][20]
                                   float* __restrict__ E)         // [50][20]
{
    const int lane = threadIdx.x & 31;
    const int wid  = threadIdx.x >> 5;   // 0..7
    const int mt   = wid & 3;            // M tile: rows mt*16 .. mt*16+15 (covers 0..63)
    const int nt   = wid >> 2;           // N tile: cols nt*16 .. nt*16+15 (covers 0..31)
    const int hi   = lane >> 4;          // 0 or 1
    const int lo   = lane & 15;

    const int mRowA = mt * 16 + lo;      // A-matrix row for this lane
    const int nCol  = nt * 16 + lo;      // B/C/D column for this lane

    v8f c = {};
    for (int k = 0; k < CHANNELS; k += 4) {
        const int kBase = k + 2 * hi;    // this lane's K pair: kBase, kBase+1
        v2f a = {};
        if (mRowA < BINS) {
            a.x = MH[mRowA * CHANNELS + kBase];
            a.y = MH[mRowA * CHANNELS + kBase + 1];
        }
        v2f b = {};
        if (nCol < EMB_SIZE) {
            b.x = W[(kBase)     * EMB_SIZE + nCol];
            b.y = W[(kBase + 1) * EMB_SIZE + nCol];
        }
        // D = A*B + C  (exec is all-1s here; divergence above has reconverged)
        c = __builtin_amdgcn_wmma_f32_16x16x4_f32(
                /*neg_a=*/false, a, /*neg_b=*/false, b,
                /*c_mod=*/(short)0, c, /*reuse_a=*/false, /*reuse_b=*/false);
    }

    if (nCol < EMB_SIZE) {
        #pragma unroll
        for (int r = 0; r < 8; ++r) {
            const int mRow = mt * 16 + hi * 8 + r;
            if (mRow < BINS) E[mRow * EMB_SIZE + nCol] = c[r];
        }
    }
}

// ---------------------------------------------------------------------------
// Kernel 2: out[b][f*20+e] = E[x[b][f]][e]  (pure gather; HBM-store bound).
// One 128-bit vector per thread: g -> (b = g/130, f = (g%130)/5, ec = g%5).
// Table lives in LDS (4 KB), read as aligned v4f. Output stored with
// non-temporal b128 (write-once stream; don't churn L2).
// ---------------------------------------------------------------------------
__global__ void mh_emb_gather(const int*   __restrict__ x,   // [B][26]
                              const float* __restrict__ E,   // [50][20]
                              v4f*         __restrict__ out, // [B*130] v4f
                              int totalVec)
{
    __shared__ __align__(16) float sE[BINS * EMB_SIZE];   // 1000 floats
    for (int i = threadIdx.x; i < BINS * EMB_SIZE; i += blockDim.x)
        sE[i] = E[i];
    __syncthreads();
    const v4f* sE4 = reinterpret_cast<const v4f*>(sE);    // [50*5]

    const int g = blockIdx.x * blockDim.x + threadIdx.x;
    if (g >= totalVec) return;

    const int b   = g / (NUM_FEATURE * 5);          // 130 v4f per batch row
    const int rem = g - b * (NUM_FEATURE * 5);
    const int f   = rem / 5;
    const int ec  = rem - f * 5;                    // which 128b chunk of the 20 floats

    const int bin = x[b * NUM_FEATURE + f];         // 5 adjacent lanes share this load
    const v4f v = sE4[bin * 5 + ec];
    __builtin_nontemporal_store(v, &out[g]);
}

extern "C" void kernel_launch(void* const* d_in, const int* in_sizes, int n_in,
                              void* d_out, int out_size, void* d_ws, size_t ws_size,
                              hipStream_t stream) {
    const int*   x  = (const int*)  d_in[0];   // [B,26] int32
    const float* W  = (const float*)d_in[1];   // [300,20] f32
    const float* MH = (const float*)d_in[2];   // [50,300] f32

    float* E = (float*)d_ws;                   // 50*20*4 = 4000 B scratch

    const int B        = in_sizes[0] / NUM_FEATURE;
    const int totalVec = B * NUM_FEATURE * (EMB_SIZE / 4); // 128-bit chunk count

    // 1) tiny WMMA GEMM builds the 50x20 embedding table (recomputed every call)
    mh_emb_build_table<<<1, 256, 0, stream>>>(MH, W, E);

    // 2) bandwidth-bound gather: one b128 NT store per thread
    const int threads = 256;
    const int blocks  = (totalVec + threads - 1) / threads;
    mh_emb_gather<<<blocks, threads, 0, stream>>>(x, E, (v4f*)d_out, totalVec);
}